// SequenceEmbedding_89575837926133
// MI455X (gfx1250) — compile-verified
//
#include <hip/hip_runtime.h>

// ext-vector types (HIP float4 struct can't feed __builtin_nontemporal_store / WMMA)
typedef __attribute__((ext_vector_type(2))) float v2f;
typedef __attribute__((ext_vector_type(4))) float v4f;
typedef __attribute__((ext_vector_type(8))) float v8f;

#define SEQ_L 4096

// 4096 blocks x 256 threads = 32768 wave32 tasks, 16 KB streamed per wave.
// half 0 (top planes, c<4):  out[c][i][j] = one_hot[i][c]  via V_WMMA_F32_16X16X4_F32
// half 1 (bottom, c>=4):     out[4+c][i][j] = one_hot[j][c] via register transpose
__global__ __launch_bounds__(256) void seq_embed_kernel(const float* __restrict__ one_hot,
                                                        float* __restrict__ out) {
  const int lane = threadIdx.x & 31;
  const int wave = (blockIdx.x << 3) | (threadIdx.x >> 5);  // 0..32767, wave-uniform
  const int half = wave >> 14;                              // 16384 tasks per half
  const int r    = wave & 16383;
  const int c    = r >> 12;                                 // channel 0..3
  const int t    = r & 4095;

  if (half == 0) {
    // ---------------- top: rank-1 broadcast via WMMA ----------------
    const int it    = t >> 4;       // i-tile 0..255 (16 rows each)
    const int chunk = t & 15;       // j-chunk 0..15 (256 cols each)
    const int i0    = it << 4;
    const int hi    = lane >> 4;    // lane half: K/M offset select
    const int m     = lane & 15;

    // A[16x4] f32 fragment: lane l, VGPR v  <->  A[m][v + 2*hi] = one_hot[i0+m][v+2*hi]
    v2f a = *(const v2f*)(one_hot + (i0 + m) * 4 + 2 * hi);
    // B[4x16] selector: B[k][j] = (k == c); lane l, VGPR v <-> K = v + 2*hi
    v2f b;
    b.x = (2 * hi + 0 == c) ? 1.0f : 0.0f;
    b.y = (2 * hi + 1 == c) ? 1.0f : 0.0f;

    v8f acc = {};
    // D[i][j] = sum_k A[i][k] * delta(k==c) = one_hot[i0+i][c]   (exact: 0/1 values)
    acc = __builtin_amdgcn_wmma_f32_16x16x4_f32(false, a, false, b, (short)0, acc,
                                                false, false);

    // D layout: lane l, VGPR v -> D[v + 8*hi][m]; value constant along j, so splat
    // to float4 and stream 256B-coalesced b128 NT stores (output >> L2, never re-read).
    float* plane = out + (long long)c * SEQ_L * SEQ_L;
    const int jbase = (chunk << 8) + (m << 2);
#pragma unroll
    for (int v = 0; v < 8; ++v) {
      const int row = i0 + v + 8 * hi;
      const float val = acc[v];
      v4f vec = {val, val, val, val};
      float* rowp = plane + (long long)row * SEQ_L + jbase;
#pragma unroll
      for (int jt = 0; jt < 4; ++jt)
        __builtin_nontemporal_store(vec, (v4f*)(rowp + jt * 64));
    }
  } else {
    // ------------- bottom: column broadcast, 4x4 register transpose -------------
    const int jchunk = t >> 7;      // 0..31  (128 cols each)
    const int ichunk = t & 127;     // 0..127 (32 rows each)
    const int j0 = (jchunk << 7) + (lane << 2);

    // rows j0..j0+3 of one_hot (contiguous b128 loads, L1-resident), pick column c
    v4f r0 = *(const v4f*)(one_hot + (j0 + 0) * 4);
    v4f r1 = *(const v4f*)(one_hot + (j0 + 1) * 4);
    v4f r2 = *(const v4f*)(one_hot + (j0 + 2) * 4);
    v4f r3 = *(const v4f*)(one_hot + (j0 + 3) * 4);
    v4f vec = {r0[c], r1[c], r2[c], r3[c]};   // = one_hot[j0..j0+3][c]

    float* plane = out + (long long)(4 + c) * SEQ_L * SEQ_L;
    const int irow0 = ichunk << 5;
#pragma unroll
    for (int i = 0; i < 32; ++i)
      __builtin_nontemporal_store(vec, (v4f*)(plane + (long long)(irow0 + i) * SEQ_L + j0));
  }
}

extern "C" void kernel_launch(void* const* d_in, const int* in_sizes, int n_in,
                              void* d_out, int out_size, void* d_ws, size_t ws_size,
                              hipStream_t stream) {
  const float* one_hot = (const float*)d_in[0];   // [4096, 4] f32
  float* out = (float*)d_out;                     // [8, 4096, 4096] f32
  (void)in_sizes; (void)n_in; (void)out_size; (void)d_ws; (void)ws_size;
  seq_embed_kernel<<<4096, 256, 0, stream>>>(one_hot, out);
}